// TrustworthinessLoss_5634997093005
// MI455X (gfx1250) — compile-verified
//
#include <hip/hip_runtime.h>
#include <math.h>

#define N 384
#define DIMX 128
#define DIMZ 32
#define KNN 5.0f

typedef __attribute__((ext_vector_type(2))) float v2f;
typedef __attribute__((ext_vector_type(8))) float v8f;

// ---------------------------------------------------------------------------
// Kernel 1: zero the output scalar, compute per-row squared norms of X and Z.
// ---------------------------------------------------------------------------
__global__ void __launch_bounds__(128) tw_prep(const float* __restrict__ X,
                                               const float* __restrict__ Z,
                                               float* __restrict__ x2,
                                               float* __restrict__ z2,
                                               float* __restrict__ out) {
  int t = blockIdx.x * blockDim.x + threadIdx.x;
  if (t == 0) out[0] = 0.0f;
  if (t < N) {
    const float* xr = X + t * DIMX;
    float sx = 0.0f;
    #pragma unroll 8
    for (int k = 0; k < DIMX; ++k) sx = fmaf(xr[k], xr[k], sx);
    x2[t] = sx;

    const float* zr = Z + t * DIMZ;
    float sz = 0.0f;
    #pragma unroll 8
    for (int k = 0; k < DIMZ; ++k) sz = fmaf(zr[k], zr[k], sz);
    z2[t] = sz;
  }
}

// ---------------------------------------------------------------------------
// Kernel 2: one wave per 16x16 tile of the Gram matrix, fp32 WMMA (16x16x4),
// K-loop unrolled x4 so loads batch ahead of the WMMA chain; distances via
// raw v_sqrt_f32 with the reference's >1e-12 guard.
// blockIdx.z == 0 -> X (kd=128), == 1 -> Z (kd=32).
//
// Fragment layout (wave32):
//   A 16x4 fp32 : lane L<16 holds (M=L, K=0..1) in v0/v1; L>=16 -> K=2..3.
//   B 4x16 fp32 : lane L<16 holds (N=L, K=0..1); L>=16 -> K=2..3.
//   C/D: elem r of lane L = (M = r + 8*(L/16), N = L%16).
// ---------------------------------------------------------------------------
__global__ void __launch_bounds__(32) tw_dist(const float* __restrict__ X,
                                              const float* __restrict__ Z,
                                              const float* __restrict__ x2,
                                              const float* __restrict__ z2,
                                              float* __restrict__ DX,
                                              float* __restrict__ DZ) {
  const bool isZ = (blockIdx.z != 0);
  const float* M   = isZ ? Z  : X;
  const float* nrm = isZ ? z2 : x2;
  float*       D   = isZ ? DZ : DX;
  const int    kd  = isZ ? DIMZ : DIMX;

  const int lane = threadIdx.x;      // 0..31, wave32, EXEC all ones
  const int half = lane >> 4;        // 0 or 1
  const int lm   = lane & 15;

  const float* rowA = M + (blockIdx.x * 16 + lm) * kd + 2 * half;
  const float* rowB = M + (blockIdx.y * 16 + lm) * kd + 2 * half;

  v8f acc = {};
  #pragma unroll 4
  for (int k0 = 0; k0 < kd; k0 += 4) {
    v2f a, b;
    a.x = rowA[k0]; a.y = rowA[k0 + 1];
    b.x = rowB[k0]; b.y = rowB[k0 + 1];
    acc = __builtin_amdgcn_wmma_f32_16x16x4_f32(
        /*neg_a=*/false, a, /*neg_b=*/false, b,
        /*c_mod=*/(short)0, acc, /*reuse_a=*/false, /*reuse_b=*/false);
  }

  const int gj = blockIdx.y * 16 + lm;
  const float nj = nrm[gj];
  #pragma unroll
  for (int r = 0; r < 8; ++r) {
    const int gi = blockIdx.x * 16 + r + 8 * half;
    float sq = fmaxf(nrm[gi] + nj - 2.0f * acc[r], 0.0f);
    D[gi * N + gj] = (sq > 1e-12f) ? __builtin_amdgcn_sqrtf(sq) : 0.0f;
  }
}

// ---------------------------------------------------------------------------
// Kernel 3: per-row soft-rank in X and Z space. Dominant n^3 work:
// sigmoid((d_b - d_a)/tau) = v_rcp_f32(1 + v_exp_f32((d_a - d_b) * 2*log2e))
// -> exactly 2 transcendental ops + mul + add per sigmoid (trans-pipe bound).
// One block (128 threads, 4 waves) per row i; rows staged in LDS.
// ---------------------------------------------------------------------------
__global__ void __launch_bounds__(128) tw_loss(const float* __restrict__ DX,
                                               const float* __restrict__ DZ,
                                               float* __restrict__ out) {
  __shared__ float rx[N];
  __shared__ float rz[N];
  __shared__ float red[128];

  const int i = blockIdx.x;
  const int t = threadIdx.x;

  for (int j = t; j < N; j += 128) {
    rx[j] = DX[i * N + j];
    rz[j] = DZ[i * N + j];
  }
  __syncthreads();

  // (d_a - d_b)/tau * log2(e) with tau = 0.5  ->  C = 2 * log2(e)
  const float C = 2.8853900817779268f;
  float lsum = 0.0f;

  for (int a = t; a < N; a += 128) {
    const float dax = rx[a];
    const float daz = rz[a];
    float sx = 0.0f, sz = 0.0f;
    #pragma unroll 4
    for (int b = 0; b < N; ++b) {
      sx += __builtin_amdgcn_rcpf(
          1.0f + __builtin_amdgcn_exp2f((dax - rx[b]) * C));
      sz += __builtin_amdgcn_rcpf(
          1.0f + __builtin_amdgcn_exp2f((daz - rz[b]) * C));
    }
    const float rank_x = 1.0f + sx;
    const float rank_z = 1.0f + sz;
    const float intrusion = fmaxf(rank_x - KNN, 0.0f);
    const float w = fminf(fmaxf((KNN + 1.0f - rank_z) * (1.0f / KNN), 0.0f), 1.0f);
    lsum += intrusion * (1.0f - w);
  }

  red[t] = lsum;
  __syncthreads();
  #pragma unroll
  for (int s = 64; s > 0; s >>= 1) {
    if (t < s) red[t] += red[t + s];
    __syncthreads();
  }
  if (t == 0) {
    // norm = 2 / (n*K*(2n - 3K - 1)) = 2 / (384*5*752)
    const float norm = 2.0f / 1443840.0f;
    atomicAdd(out, red[0] * norm);
  }
}

// ---------------------------------------------------------------------------
extern "C" void kernel_launch(void* const* d_in, const int* in_sizes, int n_in,
                              void* d_out, int out_size, void* d_ws, size_t ws_size,
                              hipStream_t stream) {
  (void)in_sizes; (void)n_in; (void)out_size; (void)ws_size;

  const float* X = (const float*)d_in[0];   // (384, 128) fp32
  const float* Z = (const float*)d_in[1];   // (384, 32)  fp32
  float* out = (float*)d_out;               // scalar fp32

  float* ws = (float*)d_ws;
  float* DX = ws;                 // 384*384
  float* DZ = ws + N * N;         // 384*384
  float* x2 = ws + 2 * N * N;     // 384
  float* z2 = x2 + N;             // 384

  tw_prep<<<dim3((N + 127) / 128), dim3(128), 0, stream>>>(X, Z, x2, z2, out);
  tw_dist<<<dim3(N / 16, N / 16, 2), dim3(32), 0, stream>>>(X, Z, x2, z2, DX, DZ);
  tw_loss<<<dim3(N), dim3(128), 0, stream>>>(DX, DZ, out);
}